// ResHGNNConv_21964462752034
// MI455X (gfx1250) — compile-verified
//
#include <hip/hip_runtime.h>

typedef __attribute__((ext_vector_type(2))) float v2f;
typedef __attribute__((ext_vector_type(8))) float v8f;

#define N_NODES 100000
#define DFT 128
#define PITCH 132   // 128 + 4 dword pad -> conflict-free ds_load_b64 across 64 banks

// ---------------------------------------------------------------------------
// Kernel 1: build CSR row pointers from sorted COO row indices (binary search)
// ---------------------------------------------------------------------------
__global__ void build_row_ptr(const int* __restrict__ row, int E,
                              int* __restrict__ row_ptr) {
    int i = blockIdx.x * blockDim.x + threadIdx.x;
    if (i > N_NODES) return;
    // lower_bound: first e with row[e] >= i  (i==N_NODES -> E)
    int lo = 0, hi = E;
    while (lo < hi) {
        int mid = (lo + hi) >> 1;
        if (row[mid] < i) lo = mid + 1; else hi = mid;
    }
    row_ptr[i] = lo;
}

// ---------------------------------------------------------------------------
// Kernel 2: fused SpMM + alpha-mix + (WMMA f32 GEMM vs W^T) + beta-mix + bias
//   8 waves / block, each wave owns one 16-row tile. Xi never touches HBM.
// ---------------------------------------------------------------------------
__global__ __launch_bounds__(256, 2)
void reshgnn_fused(const float* __restrict__ X0,
                   const int*   __restrict__ row_ptr,
                   const int*   __restrict__ col,
                   const float* __restrict__ vals,
                   const float* __restrict__ W,
                   const float* __restrict__ bias,
                   const float* __restrict__ alphap,
                   const float* __restrict__ betap,
                   float*       __restrict__ out,
                   int numTiles) {
    __shared__ float wlds[DFT * PITCH];        // W[j][k] row-major, padded
    __shared__ float xilds[8 * 16 * PITCH];    // per-wave 16x128 Xi tile, padded

    const int tid  = threadIdx.x;
    const int lane = tid & 31;
    const int wave = tid >> 5;
    const float alpha = alphap[0];
    const float beta  = betap[0];

    // ---- stage W into LDS (once per block) ----
    for (int idx = tid; idx < DFT * DFT; idx += 256) {
        int j = idx >> 7, k = idx & 127;
        wlds[j * PITCH + k] = W[idx];
    }
    __syncthreads();

    const int tile = blockIdx.x * 8 + wave;    // 16-row tile id (wave-uniform)
    if (tile < numTiles) {
        float* xw = &xilds[wave * 16 * PITCH];
        const int rowbase = tile * 16;

        // ---- SpMM + alpha residual mix: build 16x128 Xi tile in LDS ----
        for (int r = 0; r < 16; ++r) {
            const int grow = rowbase + r;                 // wave-uniform
            const int e0 = row_ptr[grow];
            const int e1 = row_ptr[grow + 1];
            float4 acc = make_float4(0.f, 0.f, 0.f, 0.f);
            for (int e = e0; e < e1; ++e) {               // uniform loop
                const int   c = col[e];                   // scalar loads
                const float v = vals[e];
                const float4 x = ((const float4*)(X0 + (long)c * DFT))[lane];
                acc.x += v * x.x; acc.y += v * x.y;
                acc.z += v * x.z; acc.w += v * x.w;
            }
            const float4 x0 = ((const float4*)(X0 + (long)grow * DFT))[lane];
            float4 xi;
            xi.x = (1.f - alpha) * acc.x + alpha * x0.x;
            xi.y = (1.f - alpha) * acc.y + alpha * x0.y;
            xi.z = (1.f - alpha) * acc.z + alpha * x0.z;
            xi.w = (1.f - alpha) * acc.w + alpha * x0.w;
            ((float4*)(xw + r * PITCH))[lane] = xi;       // 16B-aligned (528B rows)
        }
        // same-wave LDS ops are in-order: no barrier needed (tile is private)

        // ---- WMMA f32 16x16x4: out_tile = Xi(16x128) @ W^T(128x128) ----
        // A frag layout (32-bit 16x4): lanes 0-15: M=lane, K=kk*4+{0,1}
        //                              lanes 16-31: M=lane-16, K=kk*4+{2,3}
        const int m    = lane & 15;
        const int koff = (lane >> 4) << 1;                // 0 or 2

        v2f afrag[32];
        #pragma unroll
        for (int kk = 0; kk < 32; ++kk)
            afrag[kk] = *(const v2f*)(xw + m * PITCH + kk * 4 + koff);

        for (int t = 0; t < 8; ++t) {                     // 8 column tiles of 16
            const int nn = t * 16 + m;                    // output column (per lane)
            v8f acc = {};
            #pragma unroll
            for (int kk = 0; kk < 32; ++kk) {
                // B frag (4x16): lane holds W^T[k][nn] = wlds[nn][k], k=kk*4+koff+{0,1}
                v2f bfrag = *(const v2f*)(wlds + nn * PITCH + kk * 4 + koff);
                acc = __builtin_amdgcn_wmma_f32_16x16x4_f32(
                        /*neg_a=*/false, afrag[kk],
                        /*neg_b=*/false, bfrag,
                        /*c_mod=*/(short)0, acc,
                        /*reuse_a=*/false, /*reuse_b=*/false);
            }
            // ---- fused epilogue: out = (1-beta)*Xi + beta*(acc + b) ----
            // C/D layout: VGPR v -> lanes 0-15: M=v, N=lane; lanes 16-31: M=v+8
            const float bv = bias[nn];
            const int mhi = (lane >> 4) << 3;             // 0 or 8
            #pragma unroll
            for (int v = 0; v < 8; ++v) {
                const int   mrow = v + mhi;
                const float xiv  = xw[mrow * PITCH + t * 16 + m];
                const float res  = (1.f - beta) * xiv + beta * (acc[v] + bv);
                out[(long)(rowbase + mrow) * DFT + nn] = res;
            }
        }
    }
}

// ---------------------------------------------------------------------------
extern "C" void kernel_launch(void* const* d_in, const int* in_sizes, int n_in,
                              void* d_out, int out_size, void* d_ws, size_t ws_size,
                              hipStream_t stream) {
    // inputs: X(unused), X0, row, col, vals, W, b, alpha, beta
    const float* X0     = (const float*)d_in[1];
    const int*   row    = (const int*)  d_in[2];
    const int*   col    = (const int*)  d_in[3];
    const float* vals   = (const float*)d_in[4];
    const float* W      = (const float*)d_in[5];
    const float* b      = (const float*)d_in[6];
    const float* alphap = (const float*)d_in[7];
    const float* betap  = (const float*)d_in[8];
    const int E = in_sizes[2];

    int* row_ptr = (int*)d_ws;                         // (N+1) ints of scratch

    build_row_ptr<<<(N_NODES + 1 + 255) / 256, 256, 0, stream>>>(row, E, row_ptr);

    const int numTiles = (N_NODES + 15) / 16;          // 6250 (exact)
    const int grid     = (numTiles + 7) / 8;           // 8 wave-tiles per block
    reshgnn_fused<<<grid, 256, 0, stream>>>(X0, row_ptr, col, vals, W, b,
                                            alphap, betap, (float*)d_out, numTiles);
}